// HyperbolicDistanceDecoder_39127152067048
// MI455X (gfx1250) — compile-verified
//
#include <hip/hip_runtime.h>
#include <hip/hip_bf16.h>
#include <math.h>

typedef float v2f __attribute__((ext_vector_type(2)));
typedef float v8f __attribute__((ext_vector_type(8)));

constexpr int D       = 128;    // feature dim
constexpr int NROWS   = 2048;   // B*S
constexpr int KCODES  = 32000;  // codebook size
constexpr int BM      = 64;     // block tile M
constexpr int BN      = 64;     // block tile N
constexpr int LDP     = 132;    // LDS row pitch (floats): 132 -> bank = 4*row + col/..., conflict-free

// ---------------------------------------------------------------------------
// helpers
// ---------------------------------------------------------------------------
__device__ __forceinline__ float artanh_clip(float z) {
    z = fminf(fmaxf(z, -1.0f + 1e-5f), 1.0f - 1e-5f);
    return 0.5f * __logf((1.0f + z) / (1.0f - z));
}

// block-wide sum over 128 threads
__device__ __forceinline__ float bsum(float v, float* red) {
    const int t = threadIdx.x;
    red[t] = v;
    __syncthreads();
    #pragma unroll
    for (int s = 64; s > 0; s >>= 1) {
        if (t < s) red[t] += red[t + s];
        __syncthreads();
    }
    float r = red[0];
    __syncthreads();
    return r;
}

// ---------------------------------------------------------------------------
// Stage 1: per-row Poincare MLP
//   p = expmap0(v); p = mobius_add(mobius_matvec(W,p), expmap0(b));
//   x = expmap0(logmap0(p));  store x and ||x||^2
// one 128-thread block per row
// ---------------------------------------------------------------------------
__global__ __launch_bounds__(128)
void hyp_mlp_kernel(const float* __restrict__ v_in,
                    const float* __restrict__ W,
                    const float* __restrict__ bvec,
                    float* __restrict__ xg,
                    float* __restrict__ x2g) {
    __shared__ float sh_p[D];
    __shared__ float red[128];
    const int row = blockIdx.x;
    const int tid = threadIdx.x;

    // expmap0(v)
    float vi  = v_in[row * D + tid];
    float vn  = fmaxf(sqrtf(bsum(vi * vi, red)), 1e-15f);
    float pi  = tanhf(vn) / vn * vi;
    sh_p[tid] = pi;
    __syncthreads();

    // Mx = p @ W.T  (Mx[i] = sum_j W[i][j] * p[j])
    float mxi = 0.0f;
    #pragma unroll 4
    for (int j = 0; j < D; ++j) mxi += W[tid * D + j] * sh_p[j];

    // mobius_matvec scaling
    float pn  = fmaxf(sqrtf(bsum(pi * pi, red)), 1e-15f);
    float mxn = fmaxf(sqrtf(bsum(mxi * mxi, red)), 1e-15f);
    float mpi = tanhf(mxn / pn * artanh_clip(pn)) / mxn * mxi;

    // bp = expmap0(b)
    float bi  = bvec[tid];
    float bn  = fmaxf(sqrtf(bsum(bi * bi, red)), 1e-15f);
    float bpi = tanhf(bn) / bn * bi;

    // mobius_add(mp, bp)
    float X2 = bsum(mpi * mpi, red);
    float Y2 = bsum(bpi * bpi, red);
    float XY = bsum(mpi * bpi, red);
    float den = 1.0f + 2.0f * XY + X2 * Y2;
    float qi  = ((1.0f + 2.0f * XY + Y2) * mpi + (1.0f - X2) * bpi) / fmaxf(den, 1e-15f);

    // logmap0 then expmap0 (faithful, with clamps)
    float qn  = fmaxf(sqrtf(bsum(qi * qi, red)), 1e-15f);
    float v2i = artanh_clip(qn) / qn * qi;
    float v2n = fmaxf(sqrtf(bsum(v2i * v2i, red)), 1e-15f);
    float xi  = tanhf(v2n) / v2n * v2i;

    float x2 = bsum(xi * xi, red);
    xg[row * D + tid] = xi;
    if (tid == 0) x2g[row] = x2;
}

// ---------------------------------------------------------------------------
// Stage 2: fused GEMM (fp32 WMMA) + hyperbolic distance epilogue
//   one block = 64x64 output tile, 4 waves, K=128 fully staged in LDS
// ---------------------------------------------------------------------------
__global__ __launch_bounds__(128)
void hyp_dist_kernel(const float* __restrict__ xg,
                     const float* __restrict__ x2g,
                     const float* __restrict__ code,
                     const float* __restrict__ fb,
                     const float* __restrict__ temp,
                     float* __restrict__ out) {
    __shared__ float xs[BM][LDP];
    __shared__ float cs[BM][LDP];
    __shared__ float y2s[BN];
    __shared__ float fbs[BN];
    __shared__ float x2s[BM];

    const int tid = threadIdx.x;
    const int n0  = blockIdx.x * BN;
    const int m0  = blockIdx.y * BM;

    // stage x-tile [64 x 128] and code-tile [64 x 128] into LDS (float4)
    for (int idx = tid; idx < BM * (D / 4); idx += 128) {
        const int r  = idx >> 5;           // row 0..63
        const int c4 = (idx & 31) << 2;    // col 0..124 step 4
        const float4 xv = *(const float4*)(&xg[(size_t)(m0 + r) * D + c4]);
        *(float4*)(&xs[r][c4]) = xv;
        const float4 cv = *(const float4*)(&code[(size_t)(n0 + r) * D + c4]);
        *(float4*)(&cs[r][c4]) = cv;
    }
    __syncthreads();

    // stage per-row scalars: y2 (from LDS code tile), freq bias, x2
    if (tid < BN) {
        float s = 0.0f;
        #pragma unroll 4
        for (int k = 0; k < D; ++k) s += cs[tid][k] * cs[tid][k];
        y2s[tid] = s;
        fbs[tid] = fb[n0 + tid];
    }
    if (tid >= BN && tid < BN + BM) x2s[tid - BN] = x2g[m0 + (tid - BN)];
    __syncthreads();

    const int lane = tid & 31;
    const int w    = tid >> 5;   // wave id: rows [m0 + 16w, m0 + 16w + 16)
    const int l15  = lane & 15;
    const int hh   = lane >> 4;  // selects K pair within the 16x16x4 fragment
    const int ar   = w * 16 + l15;

    v8f acc[4] = {};             // 4 N-tiles of 16x16 fp32 accumulators

    // K loop: v_wmma_f32_16x16x4_f32, fragments from LDS
    for (int kk = 0; kk < D; kk += 4) {
        const int kc = kk + 2 * hh;
        v2f a;
        a.x = xs[ar][kc];
        a.y = xs[ar][kc + 1];
        #pragma unroll
        for (int j = 0; j < 4; ++j) {
            const int br = j * 16 + l15;
            v2f bvv;
            bvv.x = cs[br][kc];
            bvv.y = cs[br][kc + 1];
            acc[j] = __builtin_amdgcn_wmma_f32_16x16x4_f32(
                false, a, false, bvv, (short)0, acc[j], false, false);
        }
    }

    // epilogue: hyperbolic squared distance -> logits
    const float t      = temp[0];
    const float inv_t2 = 1.0f / (t * t);

    #pragma unroll
    for (int j = 0; j < 4; ++j) {
        const int   nl = j * 16 + l15;
        const float Y2 = y2s[nl];
        const float FB = fbs[nl];
        #pragma unroll
        for (int r = 0; r < 8; ++r) {
            const int   ml = w * 16 + r + 8 * hh;     // C layout: M = r + 8*lane[4]
            const float X2 = x2s[ml];
            const float xy = acc[j][r];
            const float aa = 1.0f - 2.0f * xy + Y2;
            const float bb = 1.0f - X2;
            const float dn = 1.0f - 2.0f * xy + X2 * Y2;
            float num2 = aa * aa * X2 - 2.0f * aa * bb * xy + bb * bb * Y2;
            float nrm  = sqrtf(fmaxf(num2, 0.0f)) / fmaxf(dn, 1e-15f);
            float z    = fminf(fmaxf(nrm, -1.0f + 1e-5f), 1.0f - 1e-5f);
            float d    = __logf((1.0f + z) / (1.0f - z));   // 2*artanh(z)
            out[(size_t)(m0 + ml) * KCODES + (n0 + nl)] = -d * d * inv_t2 + FB;
        }
    }
}

// ---------------------------------------------------------------------------
extern "C" void kernel_launch(void* const* d_in, const int* in_sizes, int n_in,
                              void* d_out, int out_size, void* d_ws, size_t ws_size,
                              hipStream_t stream) {
    (void)in_sizes; (void)n_in; (void)out_size; (void)ws_size;
    const float* v_tangent = (const float*)d_in[0];  // [16,128,128]
    const float* code      = (const float*)d_in[1];  // [32000,128]
    const float* W         = (const float*)d_in[2];  // [128,128]
    const float* bvec      = (const float*)d_in[3];  // [128]
    const float* fb        = (const float*)d_in[4];  // [32000]
    const float* temp      = (const float*)d_in[5];  // [1]
    float*       out       = (float*)d_out;          // [2048,32000]

    float* xg  = (float*)d_ws;            // [2048,128] processed points
    float* x2g = xg + (size_t)NROWS * D;  // [2048] squared norms

    hyp_mlp_kernel<<<NROWS, 128, 0, stream>>>(v_tangent, W, bvec, xg, x2g);

    dim3 grid(KCODES / BN, NROWS / BM);   // 500 x 32
    hyp_dist_kernel<<<grid, 128, 0, stream>>>(xg, x2g, code, fb, temp, out);
}